// QIDDM_A_differN_basePL_21285857919138
// MI455X (gfx1250) — compile-verified
//
#include <hip/hip_runtime.h>
#include <hip/hip_bf16.h>

// ---------------------------------------------------------------------------
// QIDDM quantum-diffusion forward for MI455X (gfx1250, wave32, WMMA).
//   PCA:    mean -> center -> Gram (WMMA f32 16x16x4, TDM-staged LDS tiles)
//           -> subspace iteration -> projection
//   Circuit: 8 statevectors per workgroup, LDS-resident; per-qubit Rot gates
//           Kronecker-fused into 16x16 unitaries applied as complex GEMMs via
//           chained V_WMMA_F32_16X16X4_F32; RZ/CZ diagonals in VALU.
// ---------------------------------------------------------------------------

typedef __attribute__((ext_vector_type(2))) float v2f;
typedef __attribute__((ext_vector_type(8))) float v8f;
typedef __attribute__((ext_vector_type(4))) unsigned int u32x4;
typedef __attribute__((ext_vector_type(4))) int i32x4;
typedef __attribute__((ext_vector_type(8))) int i32x8;

#ifndef __has_builtin
#define __has_builtin(x) 0
#endif

#if __has_builtin(__builtin_amdgcn_tensor_load_to_lds)
#define HAVE_TDM 1
#else
#define HAVE_TDM 0
#endif
#if __has_builtin(__builtin_amdgcn_s_wait_tensorcnt)
#define HAVE_TWAIT 1
#else
#define HAVE_TWAIT 0
#endif
#if __has_builtin(__builtin_amdgcn_s_cluster_barrier)
#define HAVE_CB 1
#else
#define HAVE_CB 0
#endif

#define NQ 10
#define NSTATES 1024
#define BATCH 2048
#define PIXELS 1024
#define SPB 8            // statevectors (samples) per workgroup
#define SROW 1040        // padded LDS row stride (floats) per sample

__device__ __forceinline__ v8f wmma4(v2f a, v2f b, v8f c) {
  // D(16x16 f32) = A(16x4 f32) * B(4x16 f32) + C ; chain 4x for K=16
  return __builtin_amdgcn_wmma_f32_16x16x4_f32(false, a, false, b, (short)0, c,
                                               false, false);
}

// --- Tensor Data Mover: 2D tile (f32) global -> LDS, hand-packed D# ---------
__device__ __forceinline__ void tdm_load_2d(const void* gaddr, unsigned lds_off,
                                            unsigned tensor_w, unsigned tensor_h,
                                            unsigned long long row_stride_elems,
                                            unsigned tile_w, unsigned tile_h) {
#if HAVE_TDM
  unsigned long long ga = (unsigned long long)(size_t)gaddr;
  u32x4 g0;
  g0.x = 1u;                                           // count=1 valid user D#
  g0.y = lds_off;                                      // lds_addr (bytes)
  g0.z = (unsigned)(ga & 0xffffffffull);               // global_addr[31:0]
  g0.w = (unsigned)((ga >> 32) & 0x01ffffffull) | (2u << 30); // addr[56:32]|type=2
  i32x8 g1;
  g1[0] = (int)(2u << 16);                             // data_size=4B, mask=0
  g1[1] = (int)((tensor_w & 0xffffu) << 16);           // tensor_dim0[15:0]
  g1[2] = (int)(((tensor_w >> 16) & 0xffffu) | ((tensor_h & 0xffffu) << 16));
  g1[3] = (int)(((tensor_h >> 16) & 0xffffu) | ((tile_w & 0xffffu) << 16));
  g1[4] = (int)(tile_h & 0xffffu);                     // tile_dim1, tile_dim2=0
  g1[5] = (int)(unsigned)(row_stride_elems & 0xffffffffull);
  g1[6] = (int)(unsigned)((row_stride_elems >> 32) & 0xffffull);
  g1[7] = 0;
  i32x4 z4 = {0, 0, 0, 0};
#if defined(__clang_major__) && (__clang_major__ >= 23)
  i32x8 z8 = {0, 0, 0, 0, 0, 0, 0, 0};
  __builtin_amdgcn_tensor_load_to_lds(g0, g1, z4, z4, z8, 0);
#else
  __builtin_amdgcn_tensor_load_to_lds(g0, g1, z4, z4, 0);
#endif
#else
  (void)gaddr; (void)lds_off; (void)tensor_w; (void)tensor_h;
  (void)row_stride_elems; (void)tile_w; (void)tile_h;
#endif
}

// ---------------------------------------------------------------------------
// PCA stage
// ---------------------------------------------------------------------------
__global__ void mean_kernel(const float* __restrict__ x, float* __restrict__ mean) {
  int c = blockIdx.x * blockDim.x + threadIdx.x;  // 1024 columns
  float s = 0.f;
  for (int r = 0; r < BATCH; ++r) s += x[(size_t)r * PIXELS + c];
  mean[c] = s * (1.f / (float)BATCH);
}

__global__ void center_kernel(const float* __restrict__ x,
                              const float* __restrict__ mean,
                              float* __restrict__ xc) {
  size_t i = (size_t)blockIdx.x * blockDim.x + threadIdx.x;
  xc[i] = x[i] - mean[i & (PIXELS - 1)];
}

// G = Xc^T Xc via WMMA; 64x64 output tile per block; K staged 32 rows at a
// time into LDS by the Tensor Data Mover.
__global__ void gram_kernel(const float* __restrict__ Xc, float* __restrict__ G) {
  __shared__ float la[32 * 64];
  __shared__ float lb[32 * 64];
  const int tid = threadIdx.x;
  const int lane = tid & 31;
  const int wv = tid >> 5;
  const int ln = lane & 15;
  const int kh = lane >> 4;
  const int i0 = (blockIdx.x & 15) * 64;
  const int j0 = (blockIdx.x >> 4) * 64;

  v8f acc[2] = {};
  for (int kb = 0; kb < BATCH; kb += 32) {
    __syncthreads();
#if HAVE_TDM
    if (tid == 0) {
      tdm_load_2d(Xc + (size_t)kb * PIXELS + i0, (unsigned)(size_t)(void*)la,
                  PIXELS, BATCH, (unsigned long long)PIXELS, 64u, 32u);
      tdm_load_2d(Xc + (size_t)kb * PIXELS + j0, (unsigned)(size_t)(void*)lb,
                  PIXELS, BATCH, (unsigned long long)PIXELS, 64u, 32u);
#if HAVE_TWAIT
      __builtin_amdgcn_s_wait_tensorcnt(0);
#endif
    }
#else
    for (int i = tid; i < 32 * 64; i += 256) {
      int r = i >> 6, c = i & 63;
      la[i] = Xc[(size_t)(kb + r) * PIXELS + i0 + c];
      lb[i] = Xc[(size_t)(kb + r) * PIXELS + j0 + c];
    }
#endif
    __syncthreads();
    if (kb + 32 < BATCH)
      __builtin_prefetch(Xc + (size_t)(kb + 32) * PIXELS + i0, 0, 1);
#pragma unroll
    for (int stp = 0; stp < 2; ++stp) {
      const int st = wv + stp * 8;
      const int ti = st & 3, tj = st >> 2;
#pragma unroll
      for (int kc = 0; kc < 8; ++kc) {
        const int k0 = kc * 4 + 2 * kh;
        v2f a, b;
        a.x = la[k0 * 64 + ti * 16 + ln];
        a.y = la[(k0 + 1) * 64 + ti * 16 + ln];
        b.x = lb[k0 * 64 + tj * 16 + ln];
        b.y = lb[(k0 + 1) * 64 + tj * 16 + ln];
        acc[stp] = wmma4(a, b, acc[stp]);
      }
    }
  }
#pragma unroll
  for (int stp = 0; stp < 2; ++stp) {
    const int st = wv + stp * 8;
    const int ti = st & 3, tj = st >> 2;
#pragma unroll
    for (int rr = 0; rr < 8; ++rr) {
      int row = i0 + ti * 16 + rr + 8 * kh;
      int col = j0 + tj * 16 + ln;
      G[(size_t)row * PIXELS + col] = acc[stp][rr];
    }
  }
}

__global__ void initv_kernel(float* __restrict__ V) {
  int t = blockIdx.x * blockDim.x + threadIdx.x;
  if (t < 16 * PIXELS)
    V[t] = __sinf(0.61803f * (float)(t + 1) + 0.1f * (float)(t % 97));
}

__global__ void gv_kernel(const float* __restrict__ G, const float* __restrict__ V,
                          float* __restrict__ Vn) {
  int t = blockIdx.x * blockDim.x + threadIdx.x;  // 16*1024
  int c = t >> 10, x = t & 1023;
  const float* g = G + (size_t)x * PIXELS;   // G symmetric
  const float* v = V + (size_t)c * PIXELS;
  float s = 0.f;
  for (int y = 0; y < PIXELS; ++y) s += g[y] * v[y];
  Vn[(size_t)c * PIXELS + x] = s;
}

__global__ void orthonorm_kernel(const float* __restrict__ Vin, float* __restrict__ V) {
  __shared__ float red[256];
  __shared__ float coef;
  const int tid = threadIdx.x;
  for (int i = tid; i < 16 * PIXELS; i += 256) V[i] = Vin[i];
  __syncthreads();
  for (int i = 0; i < 16; ++i) {
    float p = 0.f;
    for (int k = tid; k < PIXELS; k += 256) { float v = V[i * PIXELS + k]; p += v * v; }
    red[tid] = p; __syncthreads();
    for (int s = 128; s > 0; s >>= 1) { if (tid < s) red[tid] += red[tid + s]; __syncthreads(); }
    if (tid == 0) coef = rsqrtf(red[0] + 1e-20f);
    __syncthreads();
    float inv = coef;
    for (int k = tid; k < PIXELS; k += 256) V[i * PIXELS + k] *= inv;
    __syncthreads();
    for (int j = i + 1; j < 16; ++j) {
      float d = 0.f;
      for (int k = tid; k < PIXELS; k += 256) d += V[i * PIXELS + k] * V[j * PIXELS + k];
      red[tid] = d; __syncthreads();
      for (int s = 128; s > 0; s >>= 1) { if (tid < s) red[tid] += red[tid + s]; __syncthreads(); }
      float dd = red[0]; __syncthreads();
      for (int k = tid; k < PIXELS; k += 256) V[j * PIXELS + k] -= dd * V[i * PIXELS + k];
      __syncthreads();
    }
  }
}

__global__ void proj_kernel(const float* __restrict__ Xc, const float* __restrict__ V,
                            float* __restrict__ XR) {
  int t = blockIdx.x * blockDim.x + threadIdx.x;  // 2048*16
  int b = t >> 4, c = t & 15;
  const float* xr = Xc + (size_t)b * PIXELS;
  const float* v = V + (size_t)c * PIXELS;
  float s = 0.f;
  for (int y = 0; y < PIXELS; ++y) s += xr[y] * v[y];
  XR[(size_t)b * 16 + c] = s;
}

// ---------------------------------------------------------------------------
// Gate fusion: per sublayer build Ulow (wires 6..9, 16x16), Umid (wires 2..5,
// 16x16), Uhigh (wires 0..1, 4x4).  Layout per sublayer slot (1056 floats):
//   [0:256) ulr  [256:512) uli  [512:768) umr  [768:1024) umi
//   [1024:1040) uhr  [1040:1056) uhi
// ---------------------------------------------------------------------------
__global__ void build_unitaries_kernel(const float* __restrict__ w,
                                       float* __restrict__ U) {
  __shared__ float m2r[NQ * 4], m2i[NQ * 4];
  const int g = blockIdx.x;   // (iter*4 + layer)*2 + sub, 0..15
  const int tid = threadIdx.x;
  if (tid < NQ) {
    int base = (g * NQ + tid) * 3;
    float phi = w[base + 0], thn = w[base + 1], om = w[base + 2];
    float c = cosf(0.5f * thn), s = sinf(0.5f * thn);
    float apo = 0.5f * (phi + om), amo = 0.5f * (phi - om);
    float cpo = cosf(apo), spo = sinf(apo);
    float cmo = cosf(amo), smo = sinf(amo);
    m2r[tid * 4 + 0] =  c * cpo;  m2i[tid * 4 + 0] = -c * spo;  // u00
    m2r[tid * 4 + 1] = -s * cmo;  m2i[tid * 4 + 1] = -s * smo;  // u01
    m2r[tid * 4 + 2] =  s * cmo;  m2i[tid * 4 + 2] = -s * smo;  // u10
    m2r[tid * 4 + 3] =  c * cpo;  m2i[tid * 4 + 3] =  c * spo;  // u11
  }
  __syncthreads();
  float* outp = U + (size_t)g * 1056;
  const int a = tid >> 4, b = tid & 15;
  {  // Ulow: wires 6..9 (amp bit 3-p holds wire 6+p)
    float pr = 1.f, pi = 0.f;
#pragma unroll
    for (int p = 0; p < 4; ++p) {
      int wr = 6 + p;
      int ab = (a >> (3 - p)) & 1, bb = (b >> (3 - p)) & 1;
      float fr = m2r[wr * 4 + ab * 2 + bb], fi = m2i[wr * 4 + ab * 2 + bb];
      float nr = pr * fr - pi * fi, ni = pr * fi + pi * fr;
      pr = nr; pi = ni;
    }
    outp[tid] = pr; outp[256 + tid] = pi;
  }
  {  // Umid: wires 2..5
    float pr = 1.f, pi = 0.f;
#pragma unroll
    for (int p = 0; p < 4; ++p) {
      int wr = 2 + p;
      int ab = (a >> (3 - p)) & 1, bb = (b >> (3 - p)) & 1;
      float fr = m2r[wr * 4 + ab * 2 + bb], fi = m2i[wr * 4 + ab * 2 + bb];
      float nr = pr * fr - pi * fi, ni = pr * fi + pi * fr;
      pr = nr; pi = ni;
    }
    outp[512 + tid] = pr; outp[768 + tid] = pi;
  }
  if (tid < 16) {  // Uhigh: wires 0..1 (hi bit1 = wire0, bit0 = wire1)
    int a2 = tid >> 2, b2 = tid & 3;
    float pr = 1.f, pi = 0.f;
#pragma unroll
    for (int p = 0; p < 2; ++p) {
      int ab = (a2 >> (1 - p)) & 1, bb = (b2 >> (1 - p)) & 1;
      float fr = m2r[p * 4 + ab * 2 + bb], fi = m2i[p * 4 + ab * 2 + bb];
      float nr = pr * fr - pi * fi, ni = pr * fi + pi * fr;
      pr = nr; pi = ni;
    }
    outp[1024 + tid] = pr; outp[1040 + tid] = pi;
  }
}

// ---------------------------------------------------------------------------
// LDS-resident circuit: one diffusion iteration for SPB samples per block.
// ---------------------------------------------------------------------------
__global__ void circuit_kernel(const float* __restrict__ ang, int ang_stride,
                               const float* __restrict__ Ubase,
                               float* __restrict__ out) {
  extern __shared__ float smem[];
  float* sr  = smem;                  // SPB*SROW
  float* si  = sr + SPB * SROW;       // SPB*SROW
  float* bur = si + SPB * SROW;       // 256   (current 16x16 U, real)
  float* bui = bur + 256;             // 256   (imag)
  float* uhr = bui + 256;             // 16
  float* uhi = uhr + 16;              // 16
  float* th  = uhi + 16;              // SPB*NQ half-angles

  const int tid = threadIdx.x;
  const int lane = tid & 31;
  const int wv = tid >> 5;
  const int ln = lane & 15;
  const int kh = lane >> 4;
  const int s0 = blockIdx.x * SPB;

  for (int i = tid; i < SPB * SROW; i += 256) { sr[i] = 0.f; si[i] = 0.f; }
  __syncthreads();
  if (tid < SPB) sr[tid * SROW] = 1.f;            // |0...0>
  if (tid < SPB * NQ) {
    int s = tid / NQ, j = tid % NQ;
    // theta = (pi/2)*input ; RZ phase magnitude = theta/2 = (pi/4)*input
    th[tid] = 0.78539816339f * ang[(size_t)(s0 + s) * ang_stride + j];
  }
#if HAVE_CB
  __builtin_amdgcn_s_cluster_barrier();  // NOP when not dispatched in a cluster
#endif
  __syncthreads();

  for (int layer = 0; layer < 4; ++layer) {
    // ---- RZ data-encoding layer (diagonal phases) ----
    for (int i = tid; i < SPB * NSTATES; i += 256) {
      int s = i >> 10, a = i & 1023;
      float an = 0.f;
#pragma unroll
      for (int j = 0; j < NQ; ++j) {
        float t = th[s * NQ + j];
        an += ((a >> (9 - j)) & 1) ? t : -t;  // amp bit (9-j) <-> wire j
      }
      float sn, cs;
      __sincosf(an, &sn, &cs);
      int p = s * SROW + a;
      float r = sr[p], im = si[p];
      sr[p] = r * cs - im * sn;
      si[p] = r * sn + im * cs;
    }
    __syncthreads();

    for (int sub = 0; sub < 2; ++sub) {
      const float* ub = Ubase + (size_t)(layer * 2 + sub) * 1056;

      // ---- low-nibble fused gate (contiguous 16-vectors) ----
      bur[tid] = ub[tid];
      bui[tid] = ub[256 + tid];
      __syncthreads();
      for (int t = wv; t < 32; t += 8) {
        int R = t * 16 + ln;
        int bse = (R >> 6) * SROW + (R & 63) * 16;
        v8f cr = {}; v8f ci = {};
#pragma unroll
        for (int kc = 0; kc < 4; ++kc) {
          int k0 = kc * 4 + 2 * kh;
          v2f arr, aii, brr, bii;
          arr.x = sr[bse + k0];      arr.y = sr[bse + k0 + 1];
          aii.x = si[bse + k0];      aii.y = si[bse + k0 + 1];
          brr.x = bur[ln * 16 + k0]; brr.y = bur[ln * 16 + k0 + 1]; // B[k][n]=U[n][k]
          bii.x = bui[ln * 16 + k0]; bii.y = bui[ln * 16 + k0 + 1];
          v2f nbi = -bii;            // f32 WMMA has no A/B NEG: negate in VALU
          cr = wmma4(arr, brr, cr);  // Re += Ar*Br
          cr = wmma4(aii, nbi, cr);  // Re -= Ai*Bi
          ci = wmma4(arr, bii, ci);  // Im += Ar*Bi
          ci = wmma4(aii, brr, ci);  // Im += Ai*Br
        }
#pragma unroll
        for (int rr = 0; rr < 8; ++rr) {
          int Ro = t * 16 + rr + 8 * kh;
          int ob = (Ro >> 6) * SROW + (Ro & 63) * 16 + ln;
          sr[ob] = cr[rr];
          si[ob] = ci[rr];
        }
      }
      __syncthreads();

      // ---- mid-nibble fused gate (stride-16 16-vectors) ----
      bur[tid] = ub[512 + tid];
      bui[tid] = ub[768 + tid];
      if (tid < 16) { uhr[tid] = ub[1024 + tid]; uhi[tid] = ub[1040 + tid]; }
      __syncthreads();
      for (int t = wv; t < 32; t += 8) {
        int R = t * 16 + ln;
        int s = R >> 6, rem = R & 63;
        int bse = s * SROW + (rem >> 4) * 256 + (rem & 15);
        v8f cr = {}; v8f ci = {};
#pragma unroll
        for (int kc = 0; kc < 4; ++kc) {
          int k0 = kc * 4 + 2 * kh;
          v2f arr, aii, brr, bii;
          arr.x = sr[bse + k0 * 16];       arr.y = sr[bse + (k0 + 1) * 16];
          aii.x = si[bse + k0 * 16];       aii.y = si[bse + (k0 + 1) * 16];
          brr.x = bur[ln * 16 + k0];       brr.y = bur[ln * 16 + k0 + 1];
          bii.x = bui[ln * 16 + k0];       bii.y = bui[ln * 16 + k0 + 1];
          v2f nbi = -bii;
          cr = wmma4(arr, brr, cr);
          cr = wmma4(aii, nbi, cr);
          ci = wmma4(arr, bii, ci);
          ci = wmma4(aii, brr, ci);
        }
#pragma unroll
        for (int rr = 0; rr < 8; ++rr) {
          int Ro = t * 16 + rr + 8 * kh;
          int so = Ro >> 6, remo = Ro & 63;
          int ob = so * SROW + (remo >> 4) * 256 + (remo & 15) + ln * 16;
          sr[ob] = cr[rr];
          si[ob] = ci[rr];
        }
      }
      __syncthreads();

      // ---- high 2-qubit gate (4x4) + CZ ring, fused VALU pass ----
      const int rrot = sub + 1;  // range r = l % (n-1) + 1
      for (int q = tid; q < SPB * 256; q += 256) {
        int s = q >> 8, rest = q & 255;
        int bse = s * SROW + rest;
        float vr[4], vi[4], nr[4], ni[4];
#pragma unroll
        for (int k = 0; k < 4; ++k) { vr[k] = sr[bse + k * 256]; vi[k] = si[bse + k * 256]; }
#pragma unroll
        for (int i = 0; i < 4; ++i) {
          float ar = 0.f, ai = 0.f;
#pragma unroll
          for (int k = 0; k < 4; ++k) {
            float ur = uhr[i * 4 + k], ui = uhi[i * 4 + k];
            ar += ur * vr[k] - ui * vi[k];
            ai += ur * vi[k] + ui * vr[k];
          }
          int m10 = (i << 8) | rest;
          int rot = ((m10 >> rrot) | (m10 << (NQ - rrot))) & 1023;
          if (__popc(m10 & rot) & 1) { ar = -ar; ai = -ai; }  // CZ signs
          nr[i] = ar; ni[i] = ai;
        }
#pragma unroll
        for (int i = 0; i < 4; ++i) { sr[bse + i * 256] = nr[i]; si[bse + i * 256] = ni[i]; }
      }
      __syncthreads();
    }
  }

  // ---- probs -> clip(p*1024, 0, 1) ----
  for (int i = tid; i < SPB * NSTATES; i += 256) {
    int s = i >> 10, a = i & 1023;
    int p = s * SROW + a;
    float pr = sr[p] * sr[p] + si[p] * si[p];
    float o = fminf(fmaxf(pr * (float)PIXELS, 0.f), 1.f);
    out[(size_t)(s0 + s) * PIXELS + a] = o;
  }
}

// ---------------------------------------------------------------------------
extern "C" void kernel_launch(void* const* d_in, const int* in_sizes, int n_in,
                              void* d_out, int out_size, void* d_ws, size_t ws_size,
                              hipStream_t stream) {
  (void)in_sizes; (void)n_in; (void)out_size; (void)ws_size;
  const float* x = (const float*)d_in[0];   // (2048,1,32,32) f32
  const float* w = (const float*)d_in[1];   // (2,4,2,10,3)  f32
  float* out = (float*)d_out;
  float* ws = (float*)d_ws;

  size_t off = 0;
  float* mean = ws + off; off += 1024;
  float* Xc   = ws + off; off += (size_t)BATCH * PIXELS;
  float* G    = ws + off; off += (size_t)PIXELS * PIXELS;
  float* V    = ws + off; off += 16 * PIXELS;
  float* Vn   = ws + off; off += 16 * PIXELS;
  float* XR   = ws + off; off += (size_t)BATCH * 16;
  float* U    = ws + off; off += 16 * 1056;
  float* xrb  = ws + off; off += (size_t)BATCH * PIXELS;

  // --- PCA ---
  mean_kernel<<<PIXELS / 256, 256, 0, stream>>>(x, mean);
  center_kernel<<<(BATCH * PIXELS) / 256, 256, 0, stream>>>(x, mean, Xc);
  gram_kernel<<<256, 256, 0, stream>>>(Xc, G);
  initv_kernel<<<64, 256, 0, stream>>>(V);
  for (int it = 0; it < 8; ++it) {
    gv_kernel<<<64, 256, 0, stream>>>(G, V, Vn);
    orthonorm_kernel<<<1, 256, 0, stream>>>(Vn, V);
  }
  proj_kernel<<<(BATCH * 16) / 256, 256, 0, stream>>>(Xc, V, XR);

  // --- circuit ---
  build_unitaries_kernel<<<16, 256, 0, stream>>>(w, U);
  const size_t smem =
      (size_t)(SPB * SROW * 2 + 256 * 2 + 32 + SPB * NQ) * sizeof(float);
  circuit_kernel<<<BATCH / SPB, 256, smem, stream>>>(XR, 16, U, xrb);
  circuit_kernel<<<BATCH / SPB, 256, smem, stream>>>(xrb, PIXELS, U + 8 * 1056, out);
}